// HGTLayer_16475494547761
// MI455X (gfx1250) — compile-verified
//
#include <hip/hip_runtime.h>
#include <hip/hip_bf16.h>

#define CH 128           // IN == OUT == 128
#define NHEAD 8
// head dim 16, SQRT_DK = 4 -> scale 0.25
#define WSW_ELEMS (CH * CH)        // 16384 floats = 64KB swizzled weight matrix
#define WSW_PAIRS (WSW_ELEMS / 2)  // 8192 v2f fragments

typedef float v2f __attribute__((ext_vector_type(2)));
typedef float v8f __attribute__((ext_vector_type(8)));

// ---------------- helpers: order-preserving float<->uint for atomicMax ----
__device__ __forceinline__ unsigned f32_order_key(float f) {
    unsigned u = __float_as_uint(f);
    return (u & 0x80000000u) ? ~u : (u | 0x80000000u);
}
__device__ __forceinline__ float f32_order_decode(unsigned k) {
    return (k & 0x80000000u) ? __uint_as_float(k & 0x7FFFFFFFu)
                             : __uint_as_float(~k);
}

// ---------------- fill ----------------------------------------------------
__global__ __launch_bounds__(256) void fill_u32_kernel(unsigned* __restrict__ p,
                                                       unsigned v, int n) {
    int i = blockIdx.x * blockDim.x + threadIdx.x;
    if (i < n) p[i] = v;
}

// ---------------- swizzle raw W (row-major 128x128) into B-fragment order -
// out[((ks*8 + c)*32 + lane)*2 + j] = W[4*ks + 2*(lane>>4) + j][c*16 + (lane&15)]
__global__ __launch_bounds__(256) void swizzle_w_kernel(
    const float* __restrict__ W, float* __restrict__ Wsw)
{
    int idx = blockIdx.x * blockDim.x + threadIdx.x;
    if (idx >= WSW_PAIRS) return;
    int lane = idx & 31;
    int c    = (idx >> 5) & 7;
    int ks   = idx >> 8;
    int ka   = ks * 4 + (lane >> 4) * 2;
    int n    = c * 16 + (lane & 15);
    v2f o;
    o.x = W[ka * CH + n];
    o.y = W[(ka + 1) * CH + n];
    ((v2f*)Wsw)[idx] = o;
}

// ---------------- fold per-head relation transform, emit swizzled ---------
// WEff[i][h*16+e] = sum_d W[t][i][h*16+d] * M[r][h][d][e]  (t = r>>1)
// written directly into B-fragment order; bias rows handled by tail threads.
__global__ __launch_bounds__(256) void fold_weights_sw_kernel(
    const float* __restrict__ W,   // [2][128][128]
    const float* __restrict__ b,   // [2][128]
    const float* __restrict__ M,   // [4][8][16][16]
    float* __restrict__ WSw,       // [4][16384] swizzled
    float* __restrict__ bEff)      // [4][128]
{
    int idx = blockIdx.x * blockDim.x + threadIdx.x;
    const int nw = 4 * WSW_PAIRS;          // weight fragment pairs
    if (idx < nw) {
        int r    = idx >> 13;
        int t    = idx & (WSW_PAIRS - 1);
        int lane = t & 31;
        int c    = (t >> 5) & 7;
        int ks   = t >> 8;
        int ka   = ks * 4 + (lane >> 4) * 2;
        int e    = lane & 15;
        int h    = c;                       // col = c*16 + e  ->  h == c
        int tt   = r >> 1;
        const float* Mr = M + ((r * NHEAD + h) * 16) * 16;
        float a0 = 0.f, a1 = 0.f;
#pragma unroll
        for (int d = 0; d < 16; ++d) {
            float m = Mr[d * 16 + e];
            a0 += W[((size_t)tt * CH + ka) * CH + h * 16 + d] * m;
            a1 += W[((size_t)tt * CH + ka + 1) * CH + h * 16 + d] * m;
        }
        v2f o; o.x = a0; o.y = a1;
        ((v2f*)WSw)[idx] = o;
    } else if (idx < nw + 4 * CH) {
        int t2  = idx - nw;
        int r   = t2 >> 7;
        int col = t2 & (CH - 1);
        int h   = col >> 4, e = col & 15;
        int tt  = r >> 1;
        const float* Mr = M + ((r * NHEAD + h) * 16) * 16;
        float acc = 0.f;
#pragma unroll
        for (int d = 0; d < 16; ++d)
            acc += b[tt * CH + h * 16 + d] * Mr[d * 16 + e];
        bEff[r * CH + col] = acc;
    }
}

// ---------------- stage 64KB swizzled W into LDS --------------------------
// Uses gfx1250 async global->LDS copies (ASYNCcnt). The LDS byte address is
// passed as an asm operand (low 32 bits of the flat pointer == wave-relative
// LDS offset), which escapes the shared array so the compiler keeps the
// subsequent ds_load reads live.
__device__ __forceinline__ void stage_w_lds(float* lw, const float* __restrict__ Wsw) {
#if defined(__gfx1250__)
    unsigned lbase = (unsigned)(uintptr_t)lw + (unsigned)threadIdx.x * 16u;
    unsigned long long gbase = (unsigned long long)(uintptr_t)Wsw
                             + (unsigned long long)threadIdx.x * 16ull;
#pragma unroll
    for (int it = 0; it < 16; ++it) {           // 256 thr * 16B * 16 = 64KB
        unsigned lo = lbase + (unsigned)(it * 4096);
        unsigned long long ga = gbase + (unsigned long long)(it * 4096);
        asm volatile("global_load_async_to_lds_b128 %0, %1, off"
                     :: "v"(lo), "v"(ga) : "memory");
    }
    asm volatile("s_wait_asynccnt 0" ::: "memory");
#else
    for (int i = threadIdx.x; i < WSW_ELEMS / 4; i += 256)
        ((float4*)lw)[i] = ((const float4*)Wsw)[i];
#endif
}

// ---------------- core WMMA f32 GEMM body (W staged in LDS) ---------------
// C = A(N x 128) @ W(128 x 128) + bias ; wave computes 16 rows x 128 cols.
// A frag: lane l -> A[m][ka..ka+1], m=l&15, ka=k0+2*(l>>4)
// B frag: swizzled LDS, ds_load_b64 per column tile (bank-conflict-free)
// C/D:    VGPR r -> row r + 8*(l>>4), col n
#define GEMM_PROLOG()                                                        \
    __shared__ float lw[WSW_ELEMS];                                          \
    const int lane = threadIdx.x & 31;                                       \
    const int wave = threadIdx.x >> 5;                                       \
    const int tile = blockIdx.x * 8 + wave;                                  \
    stage_w_lds(lw, Wsw);                                                    \
    __syncthreads();                                                         \
    const int m  = lane & 15;                                                \
    const int hi = lane >> 4;

__device__ __forceinline__ void gemm_accum(const float* __restrict__ arow,
                                           const float* lw, int lane, int hi,
                                           v8f acc[8]) {
    for (int ks = 0; ks < 32; ++ks) {
        v2f a = *(const v2f*)(arow + 4 * ks + 2 * hi);
        const v2f* wb = (const v2f*)lw + (ks * 8) * 32 + lane;
#pragma unroll
        for (int c = 0; c < 8; ++c) {
            v2f bf = wb[c * 32];                    // ds_load_b64, imm offset
            acc[c] = __builtin_amdgcn_wmma_f32_16x16x4_f32(
                false, a, false, bf, (short)0, acc[c], false, false);
        }
    }
}

__global__ __launch_bounds__(256) void gemm_proj_kernel(
    const float* __restrict__ A, const float* __restrict__ Wsw,
    const float* __restrict__ bias, float* __restrict__ C, int ntiles)
{
    GEMM_PROLOG();
    if (tile >= ntiles) return;
    const float* arow = A + (size_t)(tile * 16 + m) * CH;
    v8f acc[8] = {};
    gemm_accum(arow, lw, lane, hi, acc);
    const int rbase = tile * 16 + hi * 8;
#pragma unroll
    for (int c = 0; c < 8; ++c) {
        const int col = c * 16 + m;
        const float bb = bias[col];
        float* crow = C + (size_t)rbase * CH + col;
#pragma unroll
        for (int rr = 0; rr < 8; ++rr)
            crow[(size_t)rr * CH] = acc[c][rr] + bb;
    }
}

// OUT = al * (T @ W + bias) + (1-al) * hprev ,  al = sigmoid(skip[which])
__global__ __launch_bounds__(256) void gemm_out_kernel(
    const float* __restrict__ A, const float* __restrict__ Wsw,
    const float* __restrict__ bias, const float* __restrict__ hprev,
    const float* __restrict__ skip, int which,
    float* __restrict__ OUT, int ntiles)
{
    GEMM_PROLOG();
    if (tile >= ntiles) return;
    const float al = 1.f / (1.f + __expf(-skip[which]));
    const float om = 1.f - al;
    const float* arow = A + (size_t)(tile * 16 + m) * CH;
    v8f acc[8] = {};
    gemm_accum(arow, lw, lane, hi, acc);
    const int rbase = tile * 16 + hi * 8;
#pragma unroll
    for (int c = 0; c < 8; ++c) {
        const int col = c * 16 + m;
        const float bb = bias[col];
#pragma unroll
        for (int rr = 0; rr < 8; ++rr) {
            size_t o = (size_t)(rbase + rr) * CH + col;
            OUT[o] = al * (acc[c][rr] + bb) + om * hprev[o];
        }
    }
}

// ---------------- edge pass 1: score + segment max ------------------------
__global__ __launch_bounds__(256) void edge_score_kernel(
    const float* __restrict__ q, const float* __restrict__ k,
    const int* __restrict__ src, const int* __restrict__ dst,
    const float* __restrict__ pri, float* __restrict__ score,
    unsigned* __restrict__ smax, int nEH)
{
    int idx = blockIdx.x * blockDim.x + threadIdx.x;
    if (idx >= nEH) return;
    int e = idx >> 3, h = idx & 7;
    int s = src[e], d = dst[e];
    const float4* qp = (const float4*)(q + (size_t)d * CH + h * 16);
    const float4* kp = (const float4*)(k + (size_t)s * CH + h * 16);
    float acc = 0.f;
#pragma unroll
    for (int i = 0; i < 4; ++i) {
        float4 qv = qp[i], kv = kp[i];
        acc += qv.x * kv.x + qv.y * kv.y + qv.z * kv.z + qv.w * kv.w;
    }
    float sc = acc * pri[h] * 0.25f;     // / sqrt(16)
    score[idx] = sc;
    atomicMax(&smax[d * NHEAD + h], f32_order_key(sc));
}

// ---------------- edge pass 2: exp + segment sum --------------------------
__global__ __launch_bounds__(256) void edge_expsum_kernel(
    const int* __restrict__ dst, const unsigned* __restrict__ smax,
    float* __restrict__ score, float* __restrict__ ssum, int nEH)
{
    int idx = blockIdx.x * blockDim.x + threadIdx.x;
    if (idx >= nEH) return;
    int e = idx >> 3, h = idx & 7;
    int d = dst[e];
    float mx = f32_order_decode(smax[d * NHEAD + h]);
    float ex = __expf(score[idx] - mx);
    score[idx] = ex;
    unsafeAtomicAdd(&ssum[d * NHEAD + h], ex);
}

// ---------------- edge pass 3: normalized weighted scatter of V -----------
__global__ __launch_bounds__(256) void edge_aggregate_kernel(
    const float* __restrict__ v, const int* __restrict__ src,
    const int* __restrict__ dst, const float* __restrict__ score,
    const float* __restrict__ ssum, float* __restrict__ t, int nEH)
{
    int idx = blockIdx.x * blockDim.x + threadIdx.x;
    if (idx >= nEH) return;
    int e = idx >> 3, h = idx & 7;
    int s = src[e], d = dst[e];
    float w = 0.5f * score[idx] / ssum[d * NHEAD + h];   // 0.5: cross-rel mean
    const float4* vp = (const float4*)(v + (size_t)s * CH + h * 16);
    float* tp = t + (size_t)d * CH + h * 16;
#pragma unroll
    for (int i = 0; i < 4; ++i) {
        float4 vv = vp[i];
        unsafeAtomicAdd(&tp[4 * i + 0], w * vv.x);
        unsafeAtomicAdd(&tp[4 * i + 1], w * vv.y);
        unsafeAtomicAdd(&tp[4 * i + 2], w * vv.z);
        unsafeAtomicAdd(&tp[4 * i + 3], w * vv.w);
    }
}

// ---------------- host orchestration --------------------------------------
extern "C" void kernel_launch(void* const* d_in, const int* in_sizes, int n_in,
                              void* d_out, int out_size, void* d_ws, size_t ws_size,
                              hipStream_t stream) {
    (void)n_in; (void)out_size; (void)ws_size;
    const float* h_a     = (const float*)d_in[0];
    const float* h_b     = (const float*)d_in[1];
    const float* Wk      = (const float*)d_in[2];
    const float* bk      = (const float*)d_in[3];
    const float* Wq      = (const float*)d_in[4];
    const float* bq      = (const float*)d_in[5];
    const float* Wv      = (const float*)d_in[6];
    const float* bv      = (const float*)d_in[7];
    const float* Wa      = (const float*)d_in[8];
    const float* ba      = (const float*)d_in[9];
    const float* rel_att = (const float*)d_in[10];
    const float* rel_msg = (const float*)d_in[11];
    const float* rel_pri = (const float*)d_in[12];
    const float* skip    = (const float*)d_in[13];
    const int* srcp[4] = {(const int*)d_in[14], (const int*)d_in[16],
                          (const int*)d_in[18], (const int*)d_in[20]};
    const int* dstp[4] = {(const int*)d_in[15], (const int*)d_in[17],
                          (const int*)d_in[19], (const int*)d_in[21]};

    const int N = in_sizes[0] / CH;
    const int E = in_sizes[14];
    const int ntiles = N / 16;                    // N = 20000 -> 1250
    const int gemm_blocks = (ntiles + 7) / 8;

    // ---- workspace bump allocator (256B aligned) ----
    char* wp = (char*)d_ws;
    auto walloc = [&](size_t bytes) -> char* {
        char* p = wp; wp += (bytes + 255) & ~(size_t)255; return p;
    };
    float*    qa    = (float*)walloc(sizeof(float) * (size_t)N * CH);
    float*    qb    = (float*)walloc(sizeof(float) * (size_t)N * CH);
    float*    Kbuf  = (float*)walloc(sizeof(float) * (size_t)N * CH);
    float*    Vbuf  = (float*)walloc(sizeof(float) * (size_t)N * CH);
    float*    t_a   = (float*)walloc(sizeof(float) * (size_t)N * CH);
    float*    t_b   = (float*)walloc(sizeof(float) * (size_t)N * CH);
    float*    score = (float*)walloc(sizeof(float) * (size_t)E * NHEAD);
    float*    ssum  = (float*)walloc(sizeof(float) * (size_t)N * NHEAD);
    unsigned* smax  = (unsigned*)walloc(sizeof(unsigned) * (size_t)N * NHEAD);
    float*    WkSw  = (float*)walloc(sizeof(float) * 4 * WSW_ELEMS);
    float*    WvSw  = (float*)walloc(sizeof(float) * 4 * WSW_ELEMS);
    float*    WqSw  = (float*)walloc(sizeof(float) * 2 * WSW_ELEMS);
    float*    WaSw  = (float*)walloc(sizeof(float) * 2 * WSW_ELEMS);
    float*    bkEff = (float*)walloc(sizeof(float) * 4 * CH);
    float*    bvEff = (float*)walloc(sizeof(float) * 4 * CH);

    const int nNC = N * CH;
    fill_u32_kernel<<<(nNC + 255) / 256, 256, 0, stream>>>((unsigned*)t_a, 0u, nNC);
    fill_u32_kernel<<<(nNC + 255) / 256, 256, 0, stream>>>((unsigned*)t_b, 0u, nNC);

    // fold per-head relation transforms into swizzled projection weights
    const int nfold = 4 * WSW_PAIRS + 4 * CH;
    fold_weights_sw_kernel<<<(nfold + 255) / 256, 256, 0, stream>>>(Wk, bk, rel_att, WkSw, bkEff);
    fold_weights_sw_kernel<<<(nfold + 255) / 256, 256, 0, stream>>>(Wv, bv, rel_msg, WvSw, bvEff);

    // swizzle raw Q / output weights
    const int sblocks = (WSW_PAIRS + 255) / 256;
    swizzle_w_kernel<<<sblocks, 256, 0, stream>>>(Wq,           WqSw);
    swizzle_w_kernel<<<sblocks, 256, 0, stream>>>(Wq + CH * CH, WqSw + WSW_ELEMS);
    swizzle_w_kernel<<<sblocks, 256, 0, stream>>>(Wa,           WaSw);
    swizzle_w_kernel<<<sblocks, 256, 0, stream>>>(Wa + CH * CH, WaSw + WSW_ELEMS);

    // Q projections (per destination type)
    gemm_proj_kernel<<<gemm_blocks, 256, 0, stream>>>(h_a, WqSw,             bq,      qa, ntiles);
    gemm_proj_kernel<<<gemm_blocks, 256, 0, stream>>>(h_b, WqSw + WSW_ELEMS, bq + CH, qb, ntiles);

    const int nEH = E * NHEAD;
    const int eblocks = (nEH + 255) / 256;
    const int nNH = N * NHEAD;
    for (int r = 0; r < 4; ++r) {                  // aa, ab, ba, bb
        const float* hsrc = (r < 2) ? h_a : h_b;
        const float* qd   = (r & 1) ? qb : qa;
        float*       tdst = (r & 1) ? t_b : t_a;
        gemm_proj_kernel<<<gemm_blocks, 256, 0, stream>>>(
            hsrc, WkSw + (size_t)r * WSW_ELEMS, bkEff + r * CH, Kbuf, ntiles);
        gemm_proj_kernel<<<gemm_blocks, 256, 0, stream>>>(
            hsrc, WvSw + (size_t)r * WSW_ELEMS, bvEff + r * CH, Vbuf, ntiles);
        fill_u32_kernel<<<(nNH + 255) / 256, 256, 0, stream>>>(smax, 0u, nNH);
        fill_u32_kernel<<<(nNH + 255) / 256, 256, 0, stream>>>((unsigned*)ssum, 0u, nNH);
        edge_score_kernel<<<eblocks, 256, 0, stream>>>(
            qd, Kbuf, srcp[r], dstp[r], rel_pri + r * NHEAD, score, smax, nEH);
        edge_expsum_kernel<<<eblocks, 256, 0, stream>>>(dstp[r], smax, score, ssum, nEH);
        edge_aggregate_kernel<<<eblocks, 256, 0, stream>>>(
            Vbuf, srcp[r], dstp[r], score, ssum, tdst, nEH);
    }

    float* out_a = (float*)d_out;
    float* out_b = out_a + (size_t)N * CH;
    gemm_out_kernel<<<gemm_blocks, 256, 0, stream>>>(t_a, WaSw,             ba,      h_a, skip, 0, out_a, ntiles);
    gemm_out_kernel<<<gemm_blocks, 256, 0, stream>>>(t_b, WaSw + WSW_ELEMS, ba + CH, h_b, skip, 1, out_b, ntiles);
}